// CorrectAndSmooth_48112223650272
// MI455X (gfx1250) — compile-verified
//
#include <hip/hip_runtime.h>
#include <math.h>

#define NCLS 48
#define CORR_ALPHA 0.979f
#define SMOOTH_ALPHA 0.756f
#define NUM_LAYERS 10

typedef __attribute__((ext_vector_type(2))) float v2f;
typedef __attribute__((ext_vector_type(8))) float v8f;

__device__ __forceinline__ void atomAddF(float* p, float v) {
  __hip_atomic_fetch_add(p, v, __ATOMIC_RELAXED, __HIP_MEMORY_SCOPE_AGENT);
}

__global__ void zero_kernel(float* __restrict__ p, size_t n) {
  size_t i = (size_t)blockIdx.x * blockDim.x + threadIdx.x;
  size_t stride = (size_t)gridDim.x * blockDim.x;
  for (; i < n; i += stride) p[i] = 0.0f;
}

// in-degree via hardware f32 atomics (counts are small integers -> exact in f32)
__global__ void deg_kernel(const int* __restrict__ dst, float* __restrict__ deg, int nE) {
  int e = blockIdx.x * blockDim.x + threadIdx.x;
  if (e < nE) atomAddF(&deg[dst[e]], 1.0f);
}

__global__ void norm_kernel(float* __restrict__ nrm, int n) {
  int i = blockIdx.x * blockDim.x + threadIdx.x;
  if (i < n) {
    float d = nrm[i];
    d = d < 1.0f ? 1.0f : d;
    nrm[i] = 1.0f / sqrtf(d);
  }
}

// error[train[t]] = onehot - y_soft[t]; last = (1-alpha)*error; sigma += |error|
__global__ void err_train_kernel(const float* __restrict__ ysoft,
                                 const int* __restrict__ ytrue,
                                 const int* __restrict__ train,
                                 float* __restrict__ err, float* __restrict__ last,
                                 float* __restrict__ sigma, int nTrain, float oma) {
  int idx = blockIdx.x * blockDim.x + threadIdx.x;
  float v = 0.0f;
  if (idx < nTrain * NCLS) {
    int t = idx / NCLS, c = idx % NCLS;
    float e = ((c == ytrue[t]) ? 1.0f : 0.0f) - ysoft[(size_t)t * NCLS + c];
    size_t o = (size_t)train[t] * NCLS + c;
    err[o] = e;
    last[o] = oma * e;
    v = fabsf(e);
  }
  for (int off = 16; off > 0; off >>= 1) v += __shfl_down(v, off, 32);
  if ((threadIdx.x & 31) == 0) atomAddF(sigma, v);
}

// agg[dst] += y[src] * norm[src]   (grid.y = NCLS/4 class chunks)
__global__ void edge_kernel(const int* __restrict__ src, const int* __restrict__ dst,
                            const float* __restrict__ y, const float* __restrict__ nrm,
                            float* __restrict__ agg, int nE) {
  int e = blockIdx.x * blockDim.x + threadIdx.x;
  if (e >= nE) return;
  int s = src[e], d = dst[e];
  int cb = blockIdx.y * 4;
  const float4 v = *(const float4*)(y + (size_t)s * NCLS + cb);
  float ns = nrm[s];
  float* ap = agg + (size_t)d * NCLS + cb;
  atomAddF(ap + 0, v.x * ns);
  atomAddF(ap + 1, v.y * ns);
  atomAddF(ap + 2, v.z * ns);
  atomAddF(ap + 3, v.w * ns);
}

// y = clip(last + alpha*norm*agg, lo, hi); agg = 0 for next layer
__global__ void node_kernel(float* __restrict__ y, float* __restrict__ agg,
                            const float* __restrict__ last, const float* __restrict__ nrm,
                            float alpha, float lo, float hi, int n) {
  int row = blockIdx.x * blockDim.x + threadIdx.x;
  if (row >= n) return;
  int cb = blockIdx.y * 4;
  size_t o = (size_t)row * NCLS + cb;
  float a = alpha * nrm[row];
  float4 ag = *(float4*)(agg + o);
  float4 ls = *(const float4*)(last + o);
  float4 r;
  r.x = fminf(fmaxf(ls.x + a * ag.x, lo), hi);
  r.y = fminf(fmaxf(ls.y + a * ag.y, lo), hi);
  r.z = fminf(fmaxf(ls.z + a * ag.z, lo), hi);
  r.w = fminf(fmaxf(ls.w + a * ag.w, lo), hi);
  *(float4*)(y + o) = r;
  float4 z = {0.0f, 0.0f, 0.0f, 0.0f};
  *(float4*)(agg + o) = z;
}

// scale[i] = sigma / sum_c |serr[i][c]| via V_WMMA_F32_16X16X4_F32 against all-ones B.
// One wave handles 16 rows; 12 accumulating WMMAs reduce K=48 in full fp32.
// ISA A layout (f32 16x4): lanes 0-15 hold K=0,1 ; lanes 16-31 hold K=2,3 (M = lane%16).
// B = all ones -> its layout is irrelevant. D layout: lane with N==0 (lanes 0,16)
// holds row sums for M = vgpr(+8) in acc[0..7].
__global__ void scale_wmma_kernel(const float* __restrict__ serr,
                                  const float* __restrict__ sigma,
                                  float* __restrict__ scale, int n, float invNtrain) {
  int wave = (int)((blockIdx.x * blockDim.x + threadIdx.x) >> 5);
  int lane = threadIdx.x & 31;
  int base = wave * 16;
  if (base >= n) return;                 // wave-uniform: EXEC stays all-ones for WMMA
  int row = base + (lane & 15);
  if (row >= n) row = n - 1;
  int kHalf = (lane >> 4) * 2;           // K offset for upper half-wave
  const float* rp = serr + (size_t)row * NCLS;
  v8f acc = {0.0f, 0.0f, 0.0f, 0.0f, 0.0f, 0.0f, 0.0f, 0.0f};
  v2f b = {1.0f, 1.0f};
#pragma unroll
  for (int k0 = 0; k0 < NCLS; k0 += 4) {
    v2f a;
    a.x = fabsf(rp[k0 + kHalf + 0]);
    a.y = fabsf(rp[k0 + kHalf + 1]);
    acc = __builtin_amdgcn_wmma_f32_16x16x4_f32(false, a, false, b,
                                                (short)0, acc, false, false);
  }
  float sig = sigma[0] * invNtrain;
  if ((lane & 15) == 0) {                // lanes holding column N=0
    int mb = (lane >> 4) * 8;
#pragma unroll
    for (int r = 0; r < 8; ++r) {
      int orow = base + mb + r;
      if (orow < n) {
        float s = sig / acc[r];
        if (s > 1000.0f) s = 1.0f;       // catches +inf and >1000; NaN passes through
        scale[orow] = s;
      }
    }
  }
}

__device__ __forceinline__ int cat_index(int i, const int* train, const int* valn,
                                         const int* test, int nTrain, int nVal) {
  return (i < nTrain) ? train[i]
         : (i < nTrain + nVal) ? valn[i - nTrain]
                               : test[i - nTrain - nVal];
}

// yall[cat[i]] = nan_guard(y_soft[i] + scale[i] * serr[cat[i]])
__global__ void result_kernel(const float* __restrict__ ysoft, const float* __restrict__ serr,
                              const float* __restrict__ scale,
                              const int* __restrict__ train, const int* __restrict__ valn,
                              const int* __restrict__ test, int nTrain, int nVal,
                              float* __restrict__ yall, int n) {
  int i = blockIdx.x * blockDim.x + threadIdx.x;
  if (i >= n) return;
  int node = cat_index(i, train, valn, test, nTrain, nVal);
  int cb = blockIdx.y * 4;
  float sc = scale[i];
  const float4 ys = *(const float4*)(ysoft + (size_t)i * NCLS + cb);
  const float4 se = *(const float4*)(serr + (size_t)node * NCLS + cb);
  float4 r;
  r.x = ys.x + sc * se.x; if (r.x != r.x) r.x = ys.x;
  r.y = ys.y + sc * se.y; if (r.y != r.y) r.y = ys.y;
  r.z = ys.z + sc * se.z; if (r.z != r.z) r.z = ys.z;
  r.w = ys.w + sc * se.w; if (r.w != r.w) r.w = ys.w;
  *(float4*)(yall + (size_t)node * NCLS + cb) = r;
}

__global__ void train_oh_kernel(const int* __restrict__ train, const int* __restrict__ ytrue,
                                float* __restrict__ yall, int nTrain) {
  int idx = blockIdx.x * blockDim.x + threadIdx.x;
  if (idx >= nTrain * NCLS) return;
  int t = idx / NCLS, c = idx % NCLS;
  yall[(size_t)train[t] * NCLS + c] = (c == ytrue[t]) ? 1.0f : 0.0f;
}

__global__ void prep_kernel(const float* __restrict__ y, float* __restrict__ last,
                            float oma, int n) {
  int row = blockIdx.x * blockDim.x + threadIdx.x;
  if (row >= n) return;
  int cb = blockIdx.y * 4;
  size_t o = (size_t)row * NCLS + cb;
  float4 v = *(const float4*)(y + o);
  float4 r = {oma * v.x, oma * v.y, oma * v.z, oma * v.w};
  *(float4*)(last + o) = r;
}

__global__ void gather_kernel(const float* __restrict__ yst,
                              const int* __restrict__ train, const int* __restrict__ valn,
                              const int* __restrict__ test, int nTrain, int nVal,
                              float* __restrict__ out, int n) {
  int i = blockIdx.x * blockDim.x + threadIdx.x;
  if (i >= n) return;
  int node = cat_index(i, train, valn, test, nTrain, nVal);
  int cb = blockIdx.y * 4;
  *(float4*)(out + (size_t)i * NCLS + cb) =
      *(const float4*)(yst + (size_t)node * NCLS + cb);
}

extern "C" void kernel_launch(void* const* d_in, const int* in_sizes, int n_in,
                              void* d_out, int out_size, void* d_ws, size_t ws_size,
                              hipStream_t stream) {
  (void)n_in; (void)out_size; (void)ws_size;
  const float* ysoft = (const float*)d_in[0];
  const int* ytrue = (const int*)d_in[1];
  const int* src   = (const int*)d_in[2];
  const int* dst   = (const int*)d_in[3];
  const int* train = (const int*)d_in[4];
  const int* valn  = (const int*)d_in[5];
  const int* test  = (const int*)d_in[6];
  float* out = (float*)d_out;

  const int n      = in_sizes[0] / NCLS;
  const int nE     = in_sizes[2];
  const int nTrain = in_sizes[4];
  const int nVal   = in_sizes[5];

  float* ws = (float*)d_ws;
  const size_t nc = (size_t)n * NCLS;
  float* yb    = ws;             // error -> smoothed_error (propagation state 1)
  float* yall  = yb + nc;        // corrected labels (propagation state 2)
  float* agg   = yall + nc;      // edge-scatter accumulator
  float* last  = agg + nc;       // (1-alpha)*y0 residual term
  float* nrm   = last + nc;      // deg -> deg^-0.5
  float* scal  = nrm + n;        // per-node scale
  float* sigma = scal + n;       // 1 scalar

  const int B = 256;
  const size_t totF = 4 * nc + 2 * (size_t)n + 1;

  zero_kernel<<<2048, B, 0, stream>>>(ws, totF);

  deg_kernel<<<(nE + B - 1) / B, B, 0, stream>>>(dst, nrm, nE);
  norm_kernel<<<(n + B - 1) / B, B, 0, stream>>>(nrm, n);

  err_train_kernel<<<(nTrain * NCLS + B - 1) / B, B, 0, stream>>>(
      ysoft, ytrue, train, yb, last, sigma, nTrain, 1.0f - CORR_ALPHA);

  dim3 egrid((nE + B - 1) / B, NCLS / 4);
  dim3 ngrid((n + B - 1) / B, NCLS / 4);

  // --- correct phase: 10 layers, clip [-1, 1] ---
  for (int l = 0; l < NUM_LAYERS; ++l) {
    edge_kernel<<<egrid, B, 0, stream>>>(src, dst, yb, nrm, agg, nE);
    node_kernel<<<ngrid, B, 0, stream>>>(yb, agg, last, nrm, CORR_ALPHA, -1.0f, 1.0f, n);
  }

  // per-node scale via WMMA row-abs-sum
  int nWaves = (n + 15) / 16;
  scale_wmma_kernel<<<(nWaves * 32 + B - 1) / B, B, 0, stream>>>(
      yb, sigma, scal, n, 1.0f / (float)nTrain);

  result_kernel<<<ngrid, B, 0, stream>>>(ysoft, yb, scal, train, valn, test,
                                         nTrain, nVal, yall, n);
  train_oh_kernel<<<(nTrain * NCLS + B - 1) / B, B, 0, stream>>>(train, ytrue, yall, nTrain);
  prep_kernel<<<ngrid, B, 0, stream>>>(yall, last, 1.0f - SMOOTH_ALPHA, n);

  // --- smooth phase: 10 layers, clip [0, 1] (agg already re-zeroed by node_kernel) ---
  for (int l = 0; l < NUM_LAYERS; ++l) {
    edge_kernel<<<egrid, B, 0, stream>>>(src, dst, yall, nrm, agg, nE);
    node_kernel<<<ngrid, B, 0, stream>>>(yall, agg, last, nrm, SMOOTH_ALPHA, 0.0f, 1.0f, n);
  }

  gather_kernel<<<ngrid, B, 0, stream>>>(yall, train, valn, test, nTrain, nVal, out, n);
}